// CoordinateConditioning_317827580846
// MI455X (gfx1250) — compile-verified
//
#include <hip/hip_runtime.h>
#include <hip/hip_bf16.h>
#include <math.h>

typedef __attribute__((ext_vector_type(2))) float v2f;
typedef __attribute__((ext_vector_type(8))) float v8f;

#define NATOM 2048
#define NTOK  256
#define NB    4
#define TFMR  768
#define TOKS  384
#define DIMF  256
#define NH    16

// ---------------------------------------------------------------------------
// Kernel 1: Fourier time embedding  femb[b,o] = LN(cos(2pi(t*Wf+bf))) @ Wf2s^T
// one block per batch b, 256 threads (= DIMF)
// ---------------------------------------------------------------------------
__global__ void k_femb(const float* __restrict__ times, const float* __restrict__ Wf,
                       const float* __restrict__ bf,    const float* __restrict__ g,
                       const float* __restrict__ be,    const float* __restrict__ Wf2s,
                       float* __restrict__ femb) {
  int b = blockIdx.x, f = threadIdx.x;
  __shared__ float sh[DIMF];
  __shared__ float red[DIMF];
  float t  = times[b];
  float fe = cosf(6.28318530717958647692f * (t * Wf[f] + bf[f]));
  red[f] = fe; __syncthreads();
  for (int s1 = 128; s1 > 0; s1 >>= 1) { if (f < s1) red[f] += red[f + s1]; __syncthreads(); }
  float mean = red[0] * (1.0f / DIMF); __syncthreads();
  float d0 = fe - mean;
  red[f] = d0 * d0; __syncthreads();
  for (int s1 = 128; s1 > 0; s1 >>= 1) { if (f < s1) red[f] += red[f + s1]; __syncthreads(); }
  float inv = rsqrtf(red[0] * (1.0f / DIMF) + 1e-5f); __syncthreads();
  sh[f] = d0 * inv * g[f] + be[f];
  __syncthreads();
  for (int o = f; o < TFMR; o += DIMF) {
    const float* wrow = Wf2s + (size_t)o * DIMF;
    float acc = 0.f;
    for (int k = 0; k < DIMF; ++k) acc += sh[k] * wrow[k];
    femb[b * TFMR + o] = acc;
  }
}

// ---------------------------------------------------------------------------
// Kernel 2: s = concat(s_trunk,s_inputs) @ W_single^T + femb   via WMMA f32 K=4
// M = B*T = 1024 rows, Ncols = 768.  One wave per 16x16 tile. 3072 waves.
// ---------------------------------------------------------------------------
__global__ void k_sgemm(const float* __restrict__ s_trunk, const float* __restrict__ s_inputs,
                        const float* __restrict__ W,       const float* __restrict__ femb,
                        float* __restrict__ S) {
  int wave = blockIdx.x * 8 + (threadIdx.x >> 5);
  int lane = threadIdx.x & 31;
  int tm = wave / (TFMR / 16);
  int tn = wave % (TFMR / 16);
  int m0 = tm * 16, n0 = tn * 16;
  int half = lane >> 4;
  int mr   = lane & 15;
  int arow = m0 + mr;                                  // A-matrix row for this lane
  const float* At = s_trunk + (size_t)arow * TOKS;     // row (b*T+t)*TOKS
  const float* Ai = s_inputs + (size_t)arow * TOKS;
  const float* Wrow = W + (size_t)(n0 + mr) * TFMR;    // B column n = n0+mr
  v8f acc;
  #pragma unroll
  for (int r = 0; r < 8; ++r) {
    int orow = m0 + r + 8 * half;                      // C/D: M = r + 8*(lane>=16)
    acc[r] = femb[(orow >> 8) * TFMR + n0 + mr];       // seed with time embedding
  }
  for (int k = 0; k < TFMR; k += 4) {
    int ka = k + 2 * half;                             // lanes>=16 hold K=2,3
    v2f a, bb;
    a.x  = (ka     < TOKS) ? At[ka]     : Ai[ka - TOKS];
    a.y  = (ka + 1 < TOKS) ? At[ka + 1] : Ai[ka + 1 - TOKS];
    bb.x = Wrow[ka];
    bb.y = Wrow[ka + 1];
    acc = __builtin_amdgcn_wmma_f32_16x16x4_f32(false, a, false, bb, (short)0, acc,
                                                false, false);
  }
  #pragma unroll
  for (int r = 0; r < 8; ++r) {
    int orow = m0 + r + 8 * half;
    S[(size_t)orow * TFMR + n0 + mr] = acc[r];
  }
}

// ---------------------------------------------------------------------------
// Kernel 3: S4[t] = sum_b  LN(s[b,t,:]) @ Wsc^T .  One block per token t.
// ---------------------------------------------------------------------------
__global__ void k_stoc(const float* __restrict__ S, const float* __restrict__ g,
                       const float* __restrict__ be, const float* __restrict__ Wsc,
                       float* __restrict__ S4) {
  int t = blockIdx.x, tid = threadIdx.x;
  __shared__ float red[256];
  float total = 0.f;
  for (int b = 0; b < NB; ++b) {
    const float* row = S + (size_t)(b * NTOK + t) * TFMR;
    float x0 = row[tid], x1 = row[tid + 256], x2 = row[tid + 512];
    red[tid] = x0 + x1 + x2; __syncthreads();
    for (int s1 = 128; s1 > 0; s1 >>= 1) { if (tid < s1) red[tid] += red[tid + s1]; __syncthreads(); }
    float mean = red[0] * (1.0f / TFMR); __syncthreads();
    float d0 = x0 - mean, d1 = x1 - mean, d2 = x2 - mean;
    red[tid] = d0 * d0 + d1 * d1 + d2 * d2; __syncthreads();
    for (int s1 = 128; s1 > 0; s1 >>= 1) { if (tid < s1) red[tid] += red[tid + s1]; __syncthreads(); }
    float inv = rsqrtf(red[0] * (1.0f / TFMR) + 1e-5f); __syncthreads();
    float p = (d0 * inv * g[tid      ] + be[tid      ]) * Wsc[tid      ]
            + (d1 * inv * g[tid + 256] + be[tid + 256]) * Wsc[tid + 256]
            + (d2 * inv * g[tid + 512] + be[tid + 512]) * Wsc[tid + 512];
    red[tid] = p; __syncthreads();
    for (int s1 = 128; s1 > 0; s1 >>= 1) { if (tid < s1) red[tid] += red[tid + s1]; __syncthreads(); }
    if (tid == 0) total += red[0];
    __syncthreads();
  }
  if (tid == 0) S4[t] = total;
}

// ---------------------------------------------------------------------------
// Kernel 4: Cp[n,h] = 4*(atom_feats[n]@Wa[h]+ba[h]) + S4[tok[n]];  X = sum_b x_b
// one thread per (n,h): 2048*16 threads
// ---------------------------------------------------------------------------
__global__ void k_prep(const float* __restrict__ ref_pos, const float* __restrict__ ref_charge,
                       const float* __restrict__ ref_element, const float* __restrict__ Wa,
                       const float* __restrict__ ba, const int* __restrict__ a2t,
                       const float* __restrict__ S4, const float* __restrict__ xin,
                       float* __restrict__ Cp, float* __restrict__ X) {
  int idx = blockIdx.x * 256 + threadIdx.x;
  int n = idx >> 4, h = idx & 15;
  const float* wa = Wa + h * 132;
  float c0 = ba[h];
  c0 += ref_pos[n * 3 + 0] * wa[0] + ref_pos[n * 3 + 1] * wa[1] + ref_pos[n * 3 + 2] * wa[2];
  c0 += ref_charge[n] * wa[3];
  const float* el = ref_element + (size_t)n * 128;
  float acc = 0.f;
  for (int e = 0; e < 128; ++e) acc += el[e] * wa[4 + e];
  c0 += acc;
  Cp[idx] = 4.0f * c0 + S4[a2t[n]];
  if (h < 3) {
    float x = 0.f;
    for (int b = 0; b < NB; ++b) x += xin[((size_t)b * NATOM + n) * 3 + h];
    X[n * 3 + h] = x;
  }
}

// ---------------------------------------------------------------------------
// Kernel 5 (main, bandwidth-bound): out[i,j,h] =
//   v_ij*( (X[i]-X[j])@Wc_h + Sdn_ij*Wd_h + 4*Wmask_h ) + Cp[i,h] + Cp[j,h]
// One wave handles (i, j0..j0+15): a single V_WMMA_F32_16X16X4_F32 per tile.
// A[m,:]=[Dx,Dy,Dz,Sdn] ; B=[Wc_x;Wc_y;Wc_z;Wdist]
// ---------------------------------------------------------------------------
__global__ void k_main(const float* __restrict__ xin, const float* __restrict__ X,
                       const float* __restrict__ Cp, const float* __restrict__ Wcoord,
                       const float* __restrict__ Wdist, const float* __restrict__ Wmask,
                       const int* __restrict__ pad, const int* __restrict__ uid,
                       float* __restrict__ out) {
  int wave = blockIdx.x * 8 + (threadIdx.x >> 5);
  int lane = threadIdx.x & 31;
  int i  = wave >> 7;              // N/16 = 128 j-tiles per i
  int j0 = (wave & 127) << 4;
  int half = lane >> 4;
  int m = lane & 15;               // pair index for A build; also head h for B/C/D
  int j = j0 + m;
  float Dx = X[i * 3 + 0] - X[j * 3 + 0];
  float Dy = X[i * 3 + 1] - X[j * 3 + 1];
  float Dz = X[i * 3 + 2] - X[j * 3 + 2];
  float Sd = 0.f;
  #pragma unroll
  for (int b = 0; b < NB; ++b) {
    const float* xb = xin + (size_t)b * NATOM * 3;
    float dx = xb[i * 3 + 0] - xb[j * 3 + 0];
    float dy = xb[i * 3 + 1] - xb[j * 3 + 1];
    float dz = xb[i * 3 + 2] - xb[j * 3 + 2];
    Sd += 1.0f / (1.0f + dx * dx + dy * dy + dz * dz);
  }
  v2f a, bb;
  a.x  = half ? Dz : Dx;                       // K=0/2
  a.y  = half ? Sd : Dy;                       // K=1/3
  bb.x = half ? Wcoord[m * 3 + 2] : Wcoord[m * 3 + 0];
  bb.y = half ? Wdist[m]          : Wcoord[m * 3 + 1];
  v8f acc = {};
  acc = __builtin_amdgcn_wmma_f32_16x16x4_f32(false, a, false, bb, (short)0, acc,
                                              false, false);
  int h = m;
  float wm4 = 4.0f * Wmask[h];
  float cpi = Cp[i * 16 + h];
  int padi = pad[i];
  int uidi = uid[i];
  #pragma unroll
  for (int r = 0; r < 8; ++r) {
    int mrow = r + 8 * half;                   // C/D row for this lane
    int jr = j0 + mrow;
    float v = (padi && pad[jr] && (uidi == uid[jr])) ? 1.0f : 0.0f;
    float val = v * (acc[r] + wm4) + cpi + Cp[jr * 16 + h];
    out[(((size_t)i * NATOM + jr) << 4) + h] = val;
  }
}

// ---------------------------------------------------------------------------
extern "C" void kernel_launch(void* const* d_in, const int* in_sizes, int n_in,
                              void* d_out, int out_size, void* d_ws, size_t ws_size,
                              hipStream_t stream) {
  const float* s_trunk   = (const float*)d_in[0];
  const float* s_inputs  = (const float*)d_in[1];
  const float* times     = (const float*)d_in[2];
  const float* ref_pos   = (const float*)d_in[3];
  const float* ref_chg   = (const float*)d_in[4];
  const float* ref_elem  = (const float*)d_in[5];
  const float* xnoisy    = (const float*)d_in[6];
  const float* W_single  = (const float*)d_in[7];
  const float* Wf        = (const float*)d_in[8];
  const float* bfp       = (const float*)d_in[9];
  const float* ln_f_g    = (const float*)d_in[10];
  const float* ln_f_b    = (const float*)d_in[11];
  const float* Wf2s      = (const float*)d_in[12];
  const float* Wa        = (const float*)d_in[13];
  const float* ba        = (const float*)d_in[14];
  const float* ln_s_g    = (const float*)d_in[15];
  const float* ln_s_b    = (const float*)d_in[16];
  const float* Wsc       = (const float*)d_in[17];
  const float* Wcoord    = (const float*)d_in[18];
  const float* Wdist     = (const float*)d_in[19];
  const float* Wmask     = (const float*)d_in[20];
  const int*   pad       = (const int*)d_in[21];
  const int*   uid       = (const int*)d_in[22];
  const int*   a2t       = (const int*)d_in[23];
  float* out = (float*)d_out;
  float* ws  = (float*)d_ws;

  float* femb = ws;                 // 4*768        = 3072
  float* S    = femb + 3072;        // 1024*768     = 786432
  float* S4   = S + 786432;         // 256
  float* Cp   = S4 + 256;           // 2048*16      = 32768
  float* X    = Cp + 32768;         // 2048*3       = 6144

  k_femb <<<4,     256, 0, stream>>>(times, Wf, bfp, ln_f_g, ln_f_b, Wf2s, femb);
  k_sgemm<<<384,   256, 0, stream>>>(s_trunk, s_inputs, W_single, femb, S);
  k_stoc <<<256,   256, 0, stream>>>(S, ln_s_g, ln_s_b, Wsc, S4);
  k_prep <<<128,   256, 0, stream>>>(ref_pos, ref_chg, ref_elem, Wa, ba, a2t, S4,
                                     xnoisy, Cp, X);
  k_main <<<32768, 256, 0, stream>>>(xnoisy, X, Cp, Wcoord, Wdist, Wmask, pad, uid, out);
}